// _MultiheadAttention_67645734912619
// MI455X (gfx1250) — compile-verified
//
#include <hip/hip_runtime.h>
#include <math.h>

// ---------------------------------------------------------------------------
// Problem constants (from the reference)
// ---------------------------------------------------------------------------
#define BSZ    128
#define NH     8
#define DKK    64
#define DIMM   10
#define LIN    2048
#define LOUT   2050      // conv output length (pad=2, k=3)
#define LPAD   2064      // t-padded (zero) so projection B loads are unguarded
#define CH     512       // H*DK
#define LENV   20
#define DMODEL 512

// scores-kernel tiling
#define LB     24        // l-values (K entries) per staged block
#define LBP    26        // padded per-column stride in LDS (even: aligned b64)
#define NBLK   ((LOUT + LB - 1) / LB)   // 86
#define NTILE  129       // ceil(2050/16) t-tiles for the projection band
#define PWROWS 528       // pw rows padded with zeros (c0+15 <= 524)

// dynamic LDS layout (floats) for the scores kernel
#define SM_QB   0                          // Qblk^T [512][26]
#define SM_KB   (CH * LBP)                 // 13312
#define SM_PWQ  (2 * CH * LBP)             // 26624
#define SM_PWK  (SM_PWQ + PWROWS * 12)     // +6336
#define SM_TOT  (SM_PWK + PWROWS * 12)     // 39296 floats = 157,184 B
#define SM_BYTES (SM_TOT * 4)

// ---------------------------------------------------------------------------
// CDNA5 WMMA: D(16x16,f32) += A(16x4,f32) x B(4x16,f32)
// A frag: v0/v1 = A[m][2*half+0/1], m=lane&15, half=lane>>4
// B frag: v0/v1 = B[2*half+0/1][n], n=lane&15
// D:      vgpr r = row (r + 8*half), col = lane&15
// ---------------------------------------------------------------------------
typedef __attribute__((ext_vector_type(2))) float v2f;
typedef __attribute__((ext_vector_type(8))) float v8f;

__device__ __forceinline__ v8f wmma4(v2f a, v2f b, v8f c) {
  return __builtin_amdgcn_wmma_f32_16x16x4_f32(false, a, false, b, (short)0, c,
                                               false, false);
}

// ---------------------------------------------------------------------------
// Kernel 1: depthwise conv (k=3, pad=2), stored transposed + d-padded to 12 as
// [b][t<2064][12]; rows 2050..2063 and d-slots 10,11 are ZERO so the WMMA
// projection can load unguarded 64-bit pairs.  ~26 MB -> L2-resident (192 MB).
// ---------------------------------------------------------------------------
__global__ __launch_bounds__(256) void dw_kernel(const float* __restrict__ X,
                                                 const float* __restrict__ w,
                                                 const float* __restrict__ bias,
                                                 float* __restrict__ yt) {
  int b = blockIdx.x;
  for (int t = threadIdx.x; t < LPAD; t += 256) {
    float vals[12];
#pragma unroll
    for (int d = 0; d < 12; ++d) vals[d] = 0.f;
    if (t < LOUT) {
#pragma unroll
      for (int d = 0; d < DIMM; ++d) {
        float s = bias[d];
#pragma unroll
        for (int kk = 0; kk < 3; ++kk) {
          int xi = t + kk - 2;
          if (xi >= 0 && xi < LIN)
            s = fmaf(w[d * 3 + kk], X[((size_t)b * DIMM + d) * LIN + xi], s);
        }
        vals[d] = s;
      }
    }
    float4* o = (float4*)(yt + ((size_t)b * LPAD + t) * 12);
    o[0] = make_float4(vals[0], vals[1], vals[2], vals[3]);
    o[1] = make_float4(vals[4], vals[5], vals[6], vals[7]);
    o[2] = make_float4(vals[8], vals[9], vals[10], vals[11]);
  }
}

// ---------------------------------------------------------------------------
// Kernel 2: scores + softmax, one WG (16 waves, 512 thr) per batch.
//
// Identity: with Qm = flat q buffer viewed [l=2050, j=512],
//   scores_h = Qm[:,64h:64h+64]^T * Km[:,64h:64h+64]   (K = 2050)
// and an l-block Qm[l0:l0+LB, :] is a CONTIGUOUS flat range = a contiguous
// channel band of the natural [512,2050] projection output.
//
// Per block (all loads in the hot loops are unconditional b64):
//   stage 1: WMMA projection  pw[16c x 12d] * y[12d x 16t]  (K=12, 3 steps),
//            bias rides in pw slot 10 (dead in the MAC since y slot 10 == 0),
//            scatter D into transposed LDS Qb/Kb[col][l] (stride 26).
//   stage 2: per head wave-pair (K split 12/12): fragment = ONE b64 LDS load,
//            16x v_wmma_f32_16x16x4_f32 into the 64x64 f32 accumulator.
// Epilogue: deterministic pairwise reduction in reused LDS, +prev, scale,
// scores write, 64-wide softmax, attn write.
// ---------------------------------------------------------------------------
__global__ __launch_bounds__(512) void scores_kernel(
    const float* __restrict__ ytq, const float* __restrict__ ytk,
    const float* __restrict__ pwq_g, const float* __restrict__ pbq_g,
    const float* __restrict__ pwk_g, const float* __restrict__ pbk_g,
    const float* __restrict__ prev, float* __restrict__ scores_out,
    float* __restrict__ attn_out) {
  extern __shared__ float smem[];

  int b = blockIdx.x;
  int tid = threadIdx.x;
  int wid = tid >> 5;             // 0..15
  int lane = tid & 31;
  int half = lane >> 4, ln = lane & 15;
  int head = wid & 7;             // wave pair (w, w+8) -> head w
  int pr = wid >> 3;              // 0/1: which K-half of each block

  // stage pointwise weights: [528][12], slots 0..9 = pw, slot 10 = BIAS,
  // slot 11 = 0, rows 512..527 = 0.  (y slots 10,11 are zero, so the bias in
  // slot 10 never enters the matrix product.)
  for (int c = tid; c < PWROWS; c += 512) {
    float* wq = smem + SM_PWQ + c * 12;
    float* wk = smem + SM_PWK + c * 12;
    if (c < CH) {
      const float* gq = pwq_g + c * DIMM;
      const float* gk = pwk_g + c * DIMM;
#pragma unroll
      for (int d = 0; d < DIMM; ++d) {
        wq[d] = gq[d];
        wk[d] = gk[d];
      }
      wq[10] = pbq_g[c];
      wk[10] = pbk_g[c];
      wq[11] = 0.f;
      wk[11] = 0.f;
    } else {
#pragma unroll
      for (int d = 0; d < 12; ++d) {
        wq[d] = 0.f;
        wk[d] = 0.f;
      }
    }
  }
  __syncthreads();

  const float* ytq_b = ytq + (size_t)b * LPAD * 12;
  const float* ytk_b = ytk + (size_t)b * LPAD * 12;
  int hb = head * DKK;

  v8f acc[4][4];
  v8f z = {0.f, 0.f, 0.f, 0.f, 0.f, 0.f, 0.f, 0.f};
#pragma unroll
  for (int mt = 0; mt < 4; ++mt)
#pragma unroll
    for (int nt = 0; nt < 4; ++nt) acc[mt][nt] = z;

  const float2* Qb2 = (const float2*)(smem + SM_QB);
  const float2* Kb2 = (const float2*)(smem + SM_KB);

  for (int blk = 0; blk < NBLK; ++blk) {
    int l0 = blk * LB;
    int lcount = LOUT - l0;
    if (lcount > LB) lcount = LB;
    int i0 = l0 * CH;               // flat base of this block
    int iend = i0 + lcount * CH;
    int c0 = (unsigned)i0 / LOUT;   // first channel row of the band

    // zero-fill K-slots beyond lcount (last block only) so stage 2 is
    // completely guard-free
    if (lcount < LB && tid < CH) {
      for (int ll = lcount; ll < LB; ++ll) {
        smem[SM_QB + tid * LBP + ll] = 0.f;
        smem[SM_KB + tid * LBP + ll] = 0.f;
      }
    }

    // ---- stage 1: WMMA projection into transposed Qb/Kb -------------------
    for (int tile = wid; tile < 2 * NTILE; tile += 16) {
      int which = tile & 1;                 // 0=q, 1=k
      int tbase = (tile >> 1) * 16;
      const float2* yt2 =
          (const float2*)(which ? ytk_b : ytq_b);
      const float2* pw2 =
          (const float2*)(smem + (which ? SM_PWK : SM_PWQ));
      const float* pw12 = smem + (which ? SM_PWK : SM_PWQ);
      float* dst = smem + (which ? SM_KB : SM_QB);

      int crow = c0 + ln;                   // A row (zero-padded to 528)
      int t = tbase + ln;                   // B col (yT zero-padded to 2064)
      v8f d = z;
#pragma unroll
      for (int k0 = 0; k0 < 12; k0 += 4) {  // K = 12 (10 dims + 2 zero pad)
        int kh = (k0 >> 1) + half;
        float2 a2 = pw2[crow * 6 + kh];     // ds_load_b64
        float2 b2 = yt2[t * 6 + kh];        // global_load_b64 (L2)
        v2f a, bf;
        a.x = a2.x; a.y = a2.y;
        bf.x = b2.x; bf.y = b2.y;
        d = wmma4(a, bf, d);
      }
      if (t < LOUT) {                       // scatter: (c,t) -> (l,col)
#pragma unroll
        for (int r = 0; r < 8; ++r) {
          int c = c0 + 8 * half + r;
          int i = c * LOUT + t;
          if (i >= i0 && i < iend) {
            int rel = i - i0;
            dst[(rel & 511) * LBP + (rel >> 9)] = d[r] + pw12[c * 12 + 10];
          }
        }
      }
    }
    __syncthreads();

    // ---- stage 2: scores WMMA, branch-free b64 fragments ------------------
#pragma unroll
    for (int ks = 0; ks < 12; ks += 4) {
      int lo2 = ((pr * 12 + ks) >> 1) + half;  // float2 index: always exact
      v2f Af[4], Bf[4];
#pragma unroll
      for (int mt = 0; mt < 4; ++mt) {
        int col = hb + mt * 16 + ln;
        float2 qa = Qb2[col * (LBP / 2) + lo2];
        float2 ka = Kb2[col * (LBP / 2) + lo2];
        Af[mt].x = qa.x; Af[mt].y = qa.y;
        Bf[mt].x = ka.x; Bf[mt].y = ka.y;
      }
#pragma unroll
      for (int mt = 0; mt < 4; ++mt)
#pragma unroll
        for (int nt = 0; nt < 4; ++nt)
          acc[mt][nt] = wmma4(Af[mt], Bf[nt], acc[mt][nt]);
    }
    __syncthreads();   // protect Qb/Kb before next block's projection
  }

  // ---- epilogue: pairwise reduce, +prev, scale, softmax -------------------
  float* hs = smem + head * 4160;    // reuse staging region (33,280 floats)
  if (pr == 1) {
#pragma unroll
    for (int mt = 0; mt < 4; ++mt)
#pragma unroll
      for (int nt = 0; nt < 4; ++nt)
#pragma unroll
        for (int r = 0; r < 8; ++r)
          hs[(mt * 16 + 8 * half + r) * DKK + nt * 16 + ln] = acc[mt][nt][r];
  }
  __syncthreads();
  const float* pv = prev + ((size_t)(b * NH + head)) * DKK * DKK;
  float* sco = scores_out + ((size_t)(b * NH + head)) * DKK * DKK;
  float* att = attn_out + ((size_t)(b * NH + head)) * DKK * DKK;
  if (pr == 0) {
#pragma unroll
    for (int mt = 0; mt < 4; ++mt)
#pragma unroll
      for (int nt = 0; nt < 4; ++nt)
#pragma unroll
        for (int r = 0; r < 8; ++r) {
          int m = mt * 16 + 8 * half + r;
          int n = nt * 16 + ln;
          float s = acc[mt][nt][r] + hs[m * DKK + n];
          s = s * 0.125f + pv[m * DKK + n];   // 1/sqrt(64) + realformer prev
          hs[m * DKK + n] = s;
          sco[m * DKK + n] = s;
        }
  }
  __syncthreads();
  if (pr == 0) {                     // 64-wide softmax, 2 rows per lane
#pragma unroll
    for (int rr0 = 0; rr0 < 2; ++rr0) {
      int r = lane + rr0 * 32;
      float mx = -3.4e38f;
      for (int j = 0; j < DKK; ++j) mx = fmaxf(mx, hs[r * DKK + j]);
      float sum = 0.f;
      for (int j = 0; j < DKK; ++j) sum += __expf(hs[r * DKK + j] - mx);
      float inv = 1.f / sum;
      for (int j = 0; j < DKK; ++j)
        att[r * DKK + j] = __expf(hs[r * DKK + j] - mx) * inv;
    }
  }
}

// ---------------------------------------------------------------------------
// Shared WMMA GEMM: out[20,512] = A[32,512](LDS, rows 20..31 zero) x
// Bw[n][k](global).  8 waves x 4 n-tiles, 128 K-steps; all loads b64.
// ---------------------------------------------------------------------------
__device__ __forceinline__ void gemm20(const float* __restrict__ A,
                                       const float* __restrict__ Bw,
                                       float* __restrict__ out, int tid) {
  int wave = tid >> 5, lane = tid & 31;
  int half = lane >> 4, ln = lane & 15;
  int nbase = wave * 64;
  const float2* A2 = (const float2*)A;
  const float2* B2 = (const float2*)Bw;
  v8f acc[2][4];
  v8f z = {0.f, 0.f, 0.f, 0.f, 0.f, 0.f, 0.f, 0.f};
#pragma unroll
  for (int mt = 0; mt < 2; ++mt)
#pragma unroll
    for (int nt = 0; nt < 4; ++nt) acc[mt][nt] = z;

  for (int k0 = 0; k0 < DMODEL; k0 += 4) {
    int kh = (k0 >> 1) + half;
    v2f A0, A1, Bf[4];
    float2 a0 = A2[ln * 256 + kh];
    float2 a1 = A2[(16 + ln) * 256 + kh];
    A0.x = a0.x; A0.y = a0.y;
    A1.x = a1.x; A1.y = a1.y;
#pragma unroll
    for (int nt = 0; nt < 4; ++nt) {
      int n = nbase + nt * 16 + ln;
      float2 b2 = B2[(size_t)n * 256 + kh];
      Bf[nt].x = b2.x; Bf[nt].y = b2.y;
    }
#pragma unroll
    for (int nt = 0; nt < 4; ++nt) {
      acc[0][nt] = wmma4(A0, Bf[nt], acc[0][nt]);
      acc[1][nt] = wmma4(A1, Bf[nt], acc[1][nt]);
    }
  }
#pragma unroll
  for (int mt = 0; mt < 2; ++mt)
#pragma unroll
    for (int nt = 0; nt < 4; ++nt)
#pragma unroll
      for (int r = 0; r < 8; ++r) {
        int m = mt * 16 + 8 * half + r;
        if (m < LENV)
          out[(size_t)m * DMODEL + nbase + nt * 16 + ln] = acc[mt][nt][r];
      }
}

// ---------------------------------------------------------------------------
// Kernel 3: V path. Vp = V @ vfc_w^T + b (K=10, VALU), then v = Vp @ W_V^T
// (K=512) on WMMA.  One WG per batch.  Dynamic LDS: A[32][512] + Wl + Vl.
// ---------------------------------------------------------------------------
#define VK_A   0
#define VK_WL  (32 * DMODEL)           // 16384
#define VK_VL  (VK_WL + DMODEL * DIMM) // +5120
#define VK_TOT (VK_VL + 208)
#define VK_BYTES (VK_TOT * 4)

__global__ __launch_bounds__(256) void v_kernel(const float* __restrict__ V,
                                                const float* __restrict__ vfc_w,
                                                const float* __restrict__ vfc_b,
                                                const float* __restrict__ WV,
                                                float* __restrict__ v_out) {
  extern __shared__ float sm[];
  float* Vp = sm + VK_A;
  float* Wl = sm + VK_WL;
  float* Vl = sm + VK_VL;
  int b = blockIdx.x, tid = threadIdx.x;
  for (int i = tid; i < LENV * DIMM; i += 256)
    Vl[i] = V[(size_t)b * LENV * DIMM + i];
  for (int i = tid; i < DMODEL * DIMM; i += 256) Wl[i] = vfc_w[i];
  __syncthreads();
  for (int i = tid; i < 32 * DMODEL; i += 256) {
    int l = i >> 9, dm = i & 511;
    float s = 0.f;
    if (l < LENV) {
      s = vfc_b[dm];
#pragma unroll
      for (int d = 0; d < DIMM; ++d)
        s = fmaf(Vl[l * DIMM + d], Wl[dm * DIMM + d], s);
    }
    Vp[i] = s;
  }
  __syncthreads();
  gemm20(Vp, WV, v_out + (size_t)b * LENV * DMODEL, tid);
}

// ---------------------------------------------------------------------------
// Kernel 4: context[b,h,d,t] = sum_j attn[d,j] * v[b,t,64h+j]  (64x64x20, tiny)
// ---------------------------------------------------------------------------
__global__ __launch_bounds__(256) void ctx_kernel(const float* __restrict__ attn,
                                                  const float* __restrict__ v,
                                                  float* __restrict__ ctx) {
  __shared__ float at[DKK * DKK];   // 16 KB
  __shared__ float vs[LENV * DKK];  // 5 KB
  int h = blockIdx.x, b = blockIdx.y, tid = threadIdx.x;
  for (int i = tid; i < DKK * DKK; i += 256)
    at[i] = attn[((size_t)(b * NH + h)) * DKK * DKK + i];
  for (int i = tid; i < LENV * DKK; i += 256) {
    int l = i >> 6, j = i & 63;
    vs[i] = v[((size_t)b * LENV + l) * CH + h * DKK + j];
  }
  __syncthreads();
  for (int i = tid; i < DKK * LENV; i += 256) {
    int d = i / LENV, t = i - d * LENV;
    float s = 0.f;
#pragma unroll
    for (int j = 0; j < DKK; ++j)
      s = fmaf(at[d * DKK + j], vs[t * DKK + j], s);
    ctx[(((size_t)(b * NH + h)) * DKK + d) * LENV + t] = s;
  }
}

// ---------------------------------------------------------------------------
// Kernel 5: output = ctx_resh(20x512, scrambled (d,h,t) view) @ W_O^T via WMMA.
// ---------------------------------------------------------------------------
__global__ __launch_bounds__(256) void out_kernel(const float* __restrict__ ctx,
                                                  const float* __restrict__ WO,
                                                  float* __restrict__ outp) {
  extern __shared__ float sm[];
  float* A = sm;                       // [32][512], rows 20..31 zero
  int b = blockIdx.x, tid = threadIdx.x;
  for (int i = tid; i < 32 * DMODEL; i += 256) {
    int l = i >> 9, o = i & 511;
    float val = 0.f;
    if (l < LENV) {
      int fi = l * 512 + o;            // flat (d,h,t) index
      int d = fi / 160;
      int rem = fi - d * 160;
      int hh = rem / 20;
      int tt = rem - hh * 20;
      val = ctx[(((size_t)b * NH + hh) * DKK + d) * LENV + tt];
    }
    A[i] = val;
  }
  __syncthreads();
  gemm20(A, WO, outp + (size_t)b * LENV * DMODEL, tid);
}

// ---------------------------------------------------------------------------
// Launch
// ---------------------------------------------------------------------------
extern "C" void kernel_launch(void* const* d_in, const int* in_sizes, int n_in,
                              void* d_out, int out_size, void* d_ws,
                              size_t ws_size, hipStream_t stream) {
  const float* Q    = (const float*)d_in[0];
  const float* Kin  = (const float*)d_in[1];
  const float* V    = (const float*)d_in[2];
  const float* prev = (const float*)d_in[3];
  const float* c1w  = (const float*)d_in[4];
  const float* c1b  = (const float*)d_in[5];
  const float* c1pw = (const float*)d_in[6];
  const float* c1pb = (const float*)d_in[7];
  const float* c2w  = (const float*)d_in[8];
  const float* c2b  = (const float*)d_in[9];
  const float* c2pw = (const float*)d_in[10];
  const float* c2pb = (const float*)d_in[11];
  const float* vfcw = (const float*)d_in[12];
  const float* vfcb = (const float*)d_in[13];
  const float* WV   = (const float*)d_in[14];
  const float* WO   = (const float*)d_in[15];

  // workspace: yq_t, yk_t  [B][2064][12] f32 (~26 MB, L2-resident),
  //            v [B][20][512], ctx [B][8][64][20]
  float* ws = (float*)d_ws;
  const size_t YT = (size_t)BSZ * LPAD * 12;
  float* ytq = ws;
  float* ytk = ws + YT;
  float* vws = ws + 2 * YT;
  float* ctxws = vws + (size_t)BSZ * LENV * DMODEL;

  // outputs: (output, attn, scores) concatenated
  float* outp = (float*)d_out;
  float* attn = outp + (size_t)BSZ * LENV * DMODEL;
  float* sco  = attn + (size_t)BSZ * NH * DKK * DKK;

  dw_kernel<<<dim3(BSZ), 256, 0, stream>>>(Q, c1w, c1b, ytq);
  dw_kernel<<<dim3(BSZ), 256, 0, stream>>>(Kin, c2w, c2b, ytk);
  v_kernel<<<dim3(BSZ), 256, VK_BYTES, stream>>>(V, vfcw, vfcb, WV, vws);
  scores_kernel<<<dim3(BSZ), 512, SM_BYTES, stream>>>(
      ytq, ytk, c1pw, c1pb, c2pw, c2pb, prev, sco, attn);
  ctx_kernel<<<dim3(NH, BSZ), 256, 0, stream>>>(attn, vws, ctxws);
  out_kernel<<<dim3(BSZ), 256, 32 * DMODEL * 4, stream>>>(ctxws, WO, outp);
}